// simpleLSTM_34952443855378
// MI455X (gfx1250) — compile-verified
//
#include <hip/hip_runtime.h>
#include <math.h>

typedef __attribute__((ext_vector_type(2))) float v2f;
typedef __attribute__((ext_vector_type(8))) float v8f;

#define B_    1024
#define T_    1000
#define I_    30
#define H_    128
#define G_    512
#define F1_   64
#define F2_   64
#define NCLS_ 460

// ---- LDS layout (dword offsets), bank-conflict padded for 64 banks ----
#define WHH_PITCH 520                   // pairs per k-pair row (512 + 8 pad)
#define WHH_DW    (64 * WHH_PITCH * 2)  // 66560 dw = 260 KB  (W_hh^T, K-pair interleaved)
#define H_PITCH   132                   // h/c row pitch (128 + 4 pad)
#define HC_DW     (16 * H_PITCH)        // 2112 dw
#define G_PITCH   524                   // gate row pitch (512 + 12 pad)
#define G_DW      (16 * G_PITCH)        // 8384 dw
#define XB_PITCH  34                    // x tile row pitch (30 data + 2 zero pad + 2 spare)
#define XB_BUF    (16 * XB_PITCH)       // 544 dw per buffer
#define XB_DW     (2 * XB_BUF)          // 1088 dw, double buffered
#define LDS_DW    (WHH_DW + 2 * HC_DW + G_DW + XB_DW)  // 80256 dw
#define LDS_BYTES (LDS_DW * 4)                          // 321,024 B <= 327,680 B WGP LDS

// sched_group_barrier masks
#define SGB_MFMA    0x008  // MFMA/WMMA
#define SGB_DSREAD  0x100
#define SGB_DSWRITE 0x200

__device__ __forceinline__ float sigmoidf_(float x) { return 1.0f / (1.0f + __expf(-x)); }
__device__ __forceinline__ float tanhf_(float x)    { return 1.0f - 2.0f / (__expf(2.0f * x) + 1.0f); }

__device__ __forceinline__ v8f wmma_f32(v2f a, v2f b, v8f c) {
  // V_WMMA_F32_16X16X4_F32 : D = A(16x4) * B(4x16) + C(16x16), all fp32
  return __builtin_amdgcn_wmma_f32_16x16x4_f32(false, a, false, b, (short)0, c, false, false);
}

struct Frag { v2f a, b0, b1; };

__global__ void __launch_bounds__(512, 1)
lstm_persist_wmma(const float* __restrict__ x,     // [B,T,I]
                  const float* __restrict__ W_ih,  // [4H,I]
                  const float* __restrict__ W_hh,  // [4H,H]
                  const float* __restrict__ b_ih,  // [4H]
                  const float* __restrict__ b_hh,  // [4H]
                  const float* __restrict__ W1,    // [F1,H]
                  const float* __restrict__ b1,    // [F1]
                  const float* __restrict__ W2,    // [F2,F1]
                  const float* __restrict__ b2,    // [F2]
                  const float* __restrict__ W3,    // [NCLS,F2]
                  const float* __restrict__ b3,    // [NCLS]
                  float* __restrict__ out)         // [B,NCLS]
{
  extern __shared__ float smem[];
  float* WHH = smem;                 // [64 kpairs][520 n][2]
  float* Hs  = smem + WHH_DW;        // [16][132]
  float* Cs  = Hs + HC_DW;           // [16][132]
  float* Gs  = Cs + HC_DW;           // [16][524]
  float* Xb  = Gs + G_DW;            // [2][16][34]  double-buffered x_t tile

  const int tid  = threadIdx.x;
  const int lane = tid & 31;
  const int wv   = tid >> 5;         // 16 waves
  const int hi   = lane >> 4;        // lane half selects K offset (ISA A/B layout)
  const int nn   = lane & 15;        // M index (A) / N index (B,C,D)
  const int n0   = wv * 32;          // this wave's 32-wide N strip of the 512 gates
  const int b0   = blockIdx.x * 16;  // batch tile

  // ---- stage W_hh^T into LDS as K-pair-interleaved rows: pair[k/2][n] = (W[n][k], W[n][k+1])
  for (int idx = tid; idx < G_ * H_; idx += 512) {
    int n = idx >> 7, k = idx & 127;
    WHH[(((k >> 1) * WHH_PITCH + n) << 1) | (k & 1)] = W_hh[idx];
  }
  for (int idx = tid; idx < HC_DW; idx += 512) { Hs[idx] = 0.0f; Cs[idx] = 0.0f; }

  // ---- zero the K-padding columns (30..33) of both x buffers; never rewritten
  if (tid < 128) {
    int pbuf = tid >> 6, prow = (tid >> 2) & 15, pc = 30 + (tid & 3);
    Xb[pbuf * XB_BUF + prow * XB_PITCH + pc] = 0.0f;
  }
  // ---- cooperative coalesced staging of x tile: 480 threads = 15 full waves
  const int    srow   = tid / I_;
  const int    scol   = tid - srow * I_;
  const size_t sgbase = (size_t)(b0 + srow) * (T_ * I_) + scol;
  if (tid < 480) Xb[srow * XB_PITCH + scol] = x[sgbase];   // t = 0 -> buffer 0

  // ---- loop-invariant per-lane state
  const int nA = n0 + nn, nB = nA + 16;
  const float biasA = b_ih[nA] + b_hh[nA];
  const float biasB = b_ih[nB] + b_hh[nB];

  // Hoist W_ih B-fragments (K padded 30 -> 32, pad lanes zeroed) into 32 VGPRs
  v2f bi0[8], bi1[8];
#pragma unroll
  for (int kq = 0; kq < 8; ++kq) {
    int kk = 4 * kq + 2 * hi;
    v2f t0; t0.x = 0.0f; t0.y = 0.0f;
    v2f t1; t1.x = 0.0f; t1.y = 0.0f;
    if (kk < I_) {                      // kk even: kk<=28 implies kk+1<=29 valid
      t0.x = W_ih[nA * I_ + kk]; t0.y = W_ih[nA * I_ + kk + 1];
      t1.x = W_ih[nB * I_ + kk]; t1.y = W_ih[nB * I_ + kk + 1];
    }
    bi0[kq] = t0; bi1[kq] = t1;
  }
  __syncthreads();

  // Fragment loaders (indices constant-fold under full unroll)
  auto load_xa = [&](const float* xb, int kq) -> v2f {
    int kk = 4 * kq + 2 * hi;
    return *(const v2f*)&xb[nn * XB_PITCH + kk];          // cols 30/31 are zero
  };
  auto load_hid = [&](int kq) -> Frag {
    int kk = 4 * kq + 2 * hi;
    int kp = kk >> 1;
    Frag f;
    f.a  = *(const v2f*)&Hs[nn * H_PITCH + kk];
    f.b0 = *(const v2f*)&WHH[(kp * WHH_PITCH + nA) * 2];
    f.b1 = *(const v2f*)&WHH[(kp * WHH_PITCH + nB) * 2];
    return f;
  };

  // ================= recurrence over T =================
  for (int t = 0; t < T_; ++t) {
    // Issue next step's x-tile load NOW; the 80-wmma GEMM below hides its latency.
    const bool do_stage = (t + 1 < T_) && (tid < 480);   // wave-uniform
    float xstage = 0.0f;
    if (do_stage) xstage = x[sgbase + (size_t)(t + 1) * I_];

    v8f acc0, acc1;
#pragma unroll
    for (int r = 0; r < 8; ++r) { acc0[r] = biasA; acc1[r] = biasB; }

    const float* xb = Xb + (t & 1) * XB_BUF;

    // ---- prologue: prime the software pipelines (loads drain under wmma below)
    v2f  xa0 = load_xa(xb, 0);
    v2f  xa1 = load_xa(xb, 1);
    Frag f0  = load_hid(0);
    Frag f1  = load_hid(1);

    // ---- input projection: g += x_t(16x32pad) @ W_ih^T   (16 wmma, 2-deep pipeline)
#pragma unroll
    for (int kq = 0; kq < 8; ++kq) {
      v2f xn = xa1;
      if (kq + 2 < 8) xn = load_xa(xb, kq + 2);
      acc0 = wmma_f32(xa0, bi0[kq], acc0);
      acc1 = wmma_f32(xa0, bi1[kq], acc1);
      xa0 = xa1; xa1 = xn;
    }

    // ---- hidden projection: g += h(16x128) @ W_hh^T      (64 wmma, 2-deep pipeline)
#pragma unroll
    for (int kq = 0; kq < 32; ++kq) {
      Frag fn = f1;
      if (kq + 2 < 32) fn = load_hid(kq + 2);
      acc0 = wmma_f32(f0.a, f0.b0, acc0);
      acc1 = wmma_f32(f0.a, f0.b1, acc1);
      f0 = f1; f1 = fn;
    }

    // ---- scatter gate strip to LDS (C layout: reg r -> row r + 8*half, col = nn)
#pragma unroll
    for (int r = 0; r < 8; ++r) {
      int row = r + 8 * hi;
      Gs[row * G_PITCH + nA] = acc0[r];
      Gs[row * G_PITCH + nB] = acc1[r];
    }

    // ---- scheduler steering for this block (84 DS reads, 80 WMMA, 16 DS writes):
    // burst 8 DS reads ahead, then strict [2 WMMA | 2 DS read] alternation,
    // tail WMMAs, DS writes last.  Keeps ~2 k-quads of LDS loads in flight so
    // s_wait_dscnt lands at N>0 and overlaps with matrix work.
    __builtin_amdgcn_sched_group_barrier(SGB_DSREAD, 8, 0);
#pragma unroll
    for (int i = 0; i < 40; ++i) {
      __builtin_amdgcn_sched_group_barrier(SGB_MFMA, 2, 0);
      if (i < 38) __builtin_amdgcn_sched_group_barrier(SGB_DSREAD, 2, 0);
    }
    __builtin_amdgcn_sched_group_barrier(SGB_DSWRITE, 16, 0);

    __syncthreads();

    // ---- gates + state update: wave wv owns batch row wv; lane handles 4 cols
    {
      const int m = wv;
      const int j = lane * 4;
      float4 gi = *(const float4*)&Gs[m * G_PITCH + j];
      float4 gf = *(const float4*)&Gs[m * G_PITCH + 128 + j];
      float4 gg = *(const float4*)&Gs[m * G_PITCH + 256 + j];
      float4 go = *(const float4*)&Gs[m * G_PITCH + 384 + j];
      float4 c  = *(const float4*)&Cs[m * H_PITCH + j];
      c.x = sigmoidf_(gf.x) * c.x + sigmoidf_(gi.x) * tanhf_(gg.x);
      c.y = sigmoidf_(gf.y) * c.y + sigmoidf_(gi.y) * tanhf_(gg.y);
      c.z = sigmoidf_(gf.z) * c.z + sigmoidf_(gi.z) * tanhf_(gg.z);
      c.w = sigmoidf_(gf.w) * c.w + sigmoidf_(gi.w) * tanhf_(gg.w);
      *(float4*)&Cs[m * H_PITCH + j] = c;
      float4 h;
      h.x = sigmoidf_(go.x) * tanhf_(c.x);
      h.y = sigmoidf_(go.y) * tanhf_(c.y);
      h.z = sigmoidf_(go.z) * tanhf_(c.z);
      h.w = sigmoidf_(go.w) * tanhf_(c.w);
      *(float4*)&Hs[m * H_PITCH + j] = h;
    }

    // ---- land next x tile into the other buffer (read next step, after barrier)
    if (do_stage) Xb[((t + 1) & 1) * XB_BUF + srow * XB_PITCH + scol] = xstage;
    __syncthreads();
  }

  // ================= fused MLP head on h_final (tiny) =================
  float* Z1 = Gs;            // [16][68]
  float* Z2 = Gs + 16 * 68;  // [16][68]
  for (int e = tid; e < 16 * F1_; e += 512) {
    int m = e >> 6, q = e & 63;
    float s = b1[q];
    for (int k = 0; k < H_; ++k) s += Hs[m * H_PITCH + k] * W1[q * H_ + k];
    Z1[m * 68 + q] = fmaxf(s, 0.0f);
  }
  __syncthreads();
  for (int e = tid; e < 16 * F2_; e += 512) {
    int m = e >> 6, q = e & 63;
    float s = b2[q];
    for (int k = 0; k < F1_; ++k) s += Z1[m * 68 + k] * W2[q * F1_ + k];
    Z2[m * 68 + q] = fmaxf(s, 0.0f);
  }
  __syncthreads();
  for (int e = tid; e < 16 * NCLS_; e += 512) {
    int m = e / NCLS_, q = e - m * NCLS_;
    float s = b3[q];
    for (int k = 0; k < F2_; ++k) s += Z2[m * 68 + k] * W3[q * F2_ + k];
    out[(size_t)(b0 + m) * NCLS_ + q] = s;
  }
}

extern "C" void kernel_launch(void* const* d_in, const int* in_sizes, int n_in,
                              void* d_out, int out_size, void* d_ws, size_t ws_size,
                              hipStream_t stream) {
  (void)in_sizes; (void)n_in; (void)out_size; (void)d_ws; (void)ws_size;
  const float* x    = (const float*)d_in[0];
  const float* W_ih = (const float*)d_in[1];
  const float* W_hh = (const float*)d_in[2];
  const float* b_ih = (const float*)d_in[3];
  const float* b_hh = (const float*)d_in[4];
  const float* W1   = (const float*)d_in[5];
  const float* b1   = (const float*)d_in[6];
  const float* W2   = (const float*)d_in[7];
  const float* b2   = (const float*)d_in[8];
  const float* W3   = (const float*)d_in[9];
  const float* b3   = (const float*)d_in[10];
  float* out = (float*)d_out;

  static_assert(LDS_BYTES <= 320 * 1024, "LDS budget exceeds WGP capacity");
  (void)hipFuncSetAttribute((const void*)lstm_persist_wmma,
                            hipFuncAttributeMaxDynamicSharedMemorySize, LDS_BYTES);

  dim3 grid(B_ / 16), block(512);
  lstm_persist_wmma<<<grid, block, LDS_BYTES, stream>>>(
      x, W_ih, W_hh, b_ih, b_hh, W1, b1, W2, b2, W3, b3, out);
}